// LigandGraphEncoder_21268678049993
// MI455X (gfx1250) — compile-verified
//
#include <hip/hip_runtime.h>
#include <math.h>

// Problem constants (from reference)
#define NNODES 16000
#define NEDGES 160000
#define ETOT   (NEDGES + NNODES)   // edges + self-loops
#define EMB    32
#define HID    128
#define HEADS  4
#define OUTC   (HEADS * HID)       // 512
#define NLAYER 3
#define NGRAPH 640

typedef float v2f __attribute__((ext_vector_type(2)));
typedef float v8f __attribute__((ext_vector_type(8)));

// ---------- helpers ----------

// Monotone float->uint encoding so unsigned atomicMax == float max.
__device__ __forceinline__ unsigned enc_f32(float f) {
  unsigned u = __float_as_uint(f);
  return (u & 0x80000000u) ? ~u : (u | 0x80000000u);
}
__device__ __forceinline__ float dec_f32(unsigned u) {
  return __uint_as_float((u & 0x80000000u) ? (u & 0x7FFFFFFFu) : ~u);
}
// enc(-inf) = ~0xFF800000 = 0x007FFFFF
#define ENC_NEG_INF 0x007FFFFFu

__device__ __forceinline__ void atomic_add_f32(float* p, float v) {
  __hip_atomic_fetch_add(p, v, __ATOMIC_RELAXED, __HIP_MEMORY_SCOPE_AGENT);
}

// ---------- fill kernels ----------
__global__ void fill_f32(float* __restrict__ p, float v, int n) {
  int i = blockIdx.x * blockDim.x + threadIdx.x;
  if (i < n) p[i] = v;
}
__global__ void fill_u32(unsigned* __restrict__ p, unsigned v, int n) {
  int i = blockIdx.x * blockDim.x + threadIdx.x;
  if (i < n) p[i] = v;
}

// ---------- edge_attr column mean (computed once) ----------
__global__ __launch_bounds__(256)
void eamean_kernel(const float* __restrict__ ea, float* __restrict__ out) {
  __shared__ float red[256];
  const int c = blockIdx.x;  // 0..31
  float s = 0.f;
  for (int e = threadIdx.x; e < NEDGES; e += 256)
    s += ea[(size_t)e * EMB + c];
  red[threadIdx.x] = s;
  __syncthreads();
  for (int off = 128; off > 0; off >>= 1) {
    if ((int)threadIdx.x < off) red[threadIdx.x] += red[threadIdx.x + off];
    __syncthreads();
  }
  if (threadIdx.x == 0) out[c] = red[0] * (1.0f / (float)NEDGES);
}

// ---------- WMMA GEMM: XS[N,512] = X[N,K] @ W[K,512], f32 ----------
// Block = 256 threads = 8 waves, 16 rows per block, wave w owns cols [64w,64w+64).
__global__ __launch_bounds__(256)
void gemm_xs_kernel(const float* __restrict__ X, const float* __restrict__ W,
                    float* __restrict__ XS, int K) {
  const int lane = threadIdx.x & 31;
  const int wave = threadIdx.x >> 5;
  const int m0 = blockIdx.x * 16;
  const int n0 = wave * 64;
  const int am = lane & 15;            // M index (A) / N index (B,C)
  const int ak = (lane >> 4) << 1;     // K sub-offset {0,2}

  v8f acc0 = {}, acc1 = {}, acc2 = {}, acc3 = {};
  const float* Arow = X + (size_t)(m0 + am) * K;

  for (int k = 0; k < K; k += 4) {
    v2f a;
    a.x = Arow[k + ak];
    a.y = Arow[k + ak + 1];
    const float* Bp = W + (size_t)(k + ak) * OUTC + n0 + am;
    v2f b0, b1, b2, b3;
    b0.x = Bp[0];  b0.y = Bp[OUTC];
    b1.x = Bp[16]; b1.y = Bp[OUTC + 16];
    b2.x = Bp[32]; b2.y = Bp[OUTC + 32];
    b3.x = Bp[48]; b3.y = Bp[OUTC + 48];
    acc0 = __builtin_amdgcn_wmma_f32_16x16x4_f32(false, a, false, b0, (short)0, acc0, false, false);
    acc1 = __builtin_amdgcn_wmma_f32_16x16x4_f32(false, a, false, b1, (short)0, acc1, false, false);
    acc2 = __builtin_amdgcn_wmma_f32_16x16x4_f32(false, a, false, b2, (short)0, acc2, false, false);
    acc3 = __builtin_amdgcn_wmma_f32_16x16x4_f32(false, a, false, b3, (short)0, acc3, false, false);
  }

  // C/D layout: VGPR r, lanes0-15 -> M=r, lanes16-31 -> M=r+8; N = lane&15.
  const int cn = lane & 15;
  const int cm = (lane >> 4) << 3;
#pragma unroll
  for (int r = 0; r < 8; ++r) {
    float* row = XS + (size_t)(m0 + cm + r) * OUTC + n0 + cn;
    row[0]  = acc0[r];
    row[16] = acc1[r];
    row[32] = acc2[r];
    row[48] = acc3[r];
  }
}

// ---------- per-node attention scalars: asrc/adst [N,H] ----------
__global__ void attn_node_kernel(const float* __restrict__ xs,
                                 const float* __restrict__ att_src,
                                 const float* __restrict__ att_dst,
                                 float* __restrict__ asrc, float* __restrict__ adst) {
  int idx = blockIdx.x * blockDim.x + threadIdx.x;
  if (idx >= NNODES * HEADS) return;
  const int n = idx >> 2, h = idx & 3;
  const float* row = xs + (size_t)n * OUTC + h * HID;
  const float* as = att_src + h * HID;
  const float* ad = att_dst + h * HID;
  float s = 0.f, d = 0.f;
#pragma unroll 8
  for (int c = 0; c < HID; ++c) { float v = row[c]; s += v * as[c]; d += v * ad[c]; }
  asrc[idx] = s; adst[idx] = d;
}

// ---------- collapse lin_edge: v_e[32,4] = W_e (.) att_edge ----------
__global__ void ve_kernel(const float* __restrict__ We, const float* __restrict__ att_edge,
                          float* __restrict__ ve) {
  int idx = threadIdx.x;
  if (idx >= EMB * HEADS) return;
  const int k = idx >> 2, h = idx & 3;
  float s = 0.f;
#pragma unroll 8
  for (int c = 0; c < HID; ++c)
    s += We[(size_t)k * OUTC + h * HID + c] * att_edge[h * HID + c];
  ve[k * HEADS + h] = s;
}

// ---------- edge pass 1: raw attention + leaky_relu + segment max ----------
__global__ void edge_pass1(const float* __restrict__ ea, const float* __restrict__ eamean,
                           const float* __restrict__ ve, const int* __restrict__ ei,
                           const float* __restrict__ asrc, const float* __restrict__ adst,
                           float* __restrict__ scores, unsigned* __restrict__ amax) {
  int e = blockIdx.x * blockDim.x + threadIdx.x;
  if (e >= ETOT) return;
  int s, d; const float* row;
  if (e < NEDGES) { s = ei[e]; d = ei[NEDGES + e]; row = ea + (size_t)e * EMB; }
  else            { s = e - NEDGES; d = s;          row = eamean; }
  float r[EMB];
#pragma unroll
  for (int k = 0; k < EMB; ++k) r[k] = row[k];
#pragma unroll
  for (int h = 0; h < HEADS; ++h) {
    float aeg = 0.f;
#pragma unroll
    for (int k = 0; k < EMB; ++k) aeg += r[k] * ve[k * HEADS + h];
    float a = asrc[s * HEADS + h] + adst[d * HEADS + h] + aeg;
    a = (a > 0.f) ? a : 0.2f * a;              // leaky_relu(0.2)
    scores[(size_t)e * HEADS + h] = a;
    atomicMax(&amax[d * HEADS + h], enc_f32(a));
  }
}

// ---------- edge pass 2: exp(a - max) + segment sum ----------
__global__ void edge_pass2(float* __restrict__ scores, const int* __restrict__ ei,
                           const unsigned* __restrict__ amax, float* __restrict__ den) {
  int idx = blockIdx.x * blockDim.x + threadIdx.x;
  if (idx >= ETOT * HEADS) return;
  const int e = idx >> 2, h = idx & 3;
  const int d = (e < NEDGES) ? ei[NEDGES + e] : (e - NEDGES);
  float m = dec_f32(amax[d * HEADS + h]);
  float w = __expf(scores[idx] - m);
  scores[idx] = w;
  atomic_add_f32(&den[d * HEADS + h], w);
}

// ---------- scatter aggregation: agg[dst] += xs[src] * coeff (wave per edge) ----------
__global__ __launch_bounds__(256)
void scatter_kernel(const float* __restrict__ xs, const float* __restrict__ scores,
                    const float* __restrict__ den, const int* __restrict__ ei,
                    float* __restrict__ agg) {
  const int gid = blockIdx.x * blockDim.x + threadIdx.x;
  const int e = gid >> 5;
  const int lane = gid & 31;
  if (e >= ETOT) return;
  int s, d;
  if (e < NEDGES) { s = ei[e]; d = ei[NEDGES + e]; }
  else            { s = e - NEDGES; d = s; }
  float coeff[HEADS];
#pragma unroll
  for (int h = 0; h < HEADS; ++h)
    coeff[h] = scores[(size_t)e * HEADS + h] / (den[d * HEADS + h] + 1e-16f);
  const float* xrow = xs + (size_t)s * OUTC;
  float* orow = agg + (size_t)d * OUTC;
#pragma unroll
  for (int j = lane; j < OUTC; j += 32)
    atomic_add_f32(&orow[j], xrow[j] * coeff[j >> 7]);
}

// ---------- head-mean + bias + LayerNorm + ReLU ----------
__global__ __launch_bounds__(128)
void finalize_kernel(const float* __restrict__ agg, const float* __restrict__ bias,
                     const float* __restrict__ g, const float* __restrict__ b,
                     float* __restrict__ xout) {
  __shared__ float red[128];
  const int n = blockIdx.x, c = threadIdx.x;
  const float* row = agg + (size_t)n * OUTC;
  float v = 0.25f * (row[c] + row[HID + c] + row[2 * HID + c] + row[3 * HID + c]) + bias[c];
  red[c] = v; __syncthreads();
  for (int off = 64; off > 0; off >>= 1) {
    if (c < off) red[c] += red[c + off];
    __syncthreads();
  }
  float mu = red[0] * (1.0f / HID);
  __syncthreads();
  float dv = v - mu;
  red[c] = dv * dv; __syncthreads();
  for (int off = 64; off > 0; off >>= 1) {
    if (c < off) red[c] += red[c + off];
    __syncthreads();
  }
  float var = red[0] * (1.0f / HID);
  float y = dv * rsqrtf(var + 1e-5f) * g[c] + b[c];
  xout[(size_t)n * HID + c] = fmaxf(y, 0.f);
}

// ---------- global mean pool ----------
__global__ void pool_accum(const float* __restrict__ x, const int* __restrict__ batch,
                           float* __restrict__ sums, float* __restrict__ cnt) {
  int idx = blockIdx.x * blockDim.x + threadIdx.x;
  if (idx >= NNODES * HID) return;
  const int n = idx >> 7, c = idx & 127;
  const int gidx = batch[n];
  atomic_add_f32(&sums[(size_t)gidx * HID + c], x[idx]);
  if (c == 0) atomic_add_f32(&cnt[gidx], 1.0f);
}
__global__ void pool_div(float* __restrict__ sums, const float* __restrict__ cnt) {
  int idx = blockIdx.x * blockDim.x + threadIdx.x;
  if (idx >= NGRAPH * HID) return;
  sums[idx] /= fmaxf(cnt[idx >> 7], 1.0f);
}

// ---------- launch ----------
extern "C" void kernel_launch(void* const* d_in, const int* in_sizes, int n_in,
                              void* d_out, int out_size, void* d_ws, size_t ws_size,
                              hipStream_t stream) {
  (void)in_sizes; (void)n_in; (void)out_size; (void)ws_size;

  const float* x_in      = (const float*)d_in[0];
  const float* edge_attr = (const float*)d_in[1];
  const int*   ei        = (const int*)d_in[2];   // [2,E]: src then dst
  const int*   batch     = (const int*)d_in[3];

  char* ws = (char*)d_ws;
  size_t off = 0;
  auto alloc = [&](size_t bytes) -> void* {
    void* p = ws + off;
    off = (off + bytes + 255) & ~(size_t)255;
    return p;
  };
  float*    xs     = (float*)   alloc((size_t)NNODES * OUTC * 4);   // 32.8 MB
  float*    agg    = (float*)   alloc((size_t)NNODES * OUTC * 4);   // 32.8 MB
  float*    sc     = (float*)   alloc((size_t)ETOT * HEADS * 4);    // 2.8 MB
  unsigned* amax   = (unsigned*)alloc((size_t)NNODES * HEADS * 4);
  float*    den    = (float*)   alloc((size_t)NNODES * HEADS * 4);
  float*    asrc   = (float*)   alloc((size_t)NNODES * HEADS * 4);
  float*    adst   = (float*)   alloc((size_t)NNODES * HEADS * 4);
  float*    xcur   = (float*)   alloc((size_t)NNODES * HID * 4);    // 8.2 MB
  float*    eamean = (float*)   alloc(EMB * 4);
  float*    ve     = (float*)   alloc(EMB * HEADS * 4);
  float*    cnt    = (float*)   alloc(NGRAPH * 4);

  eamean_kernel<<<EMB, 256, 0, stream>>>(edge_attr, eamean);

  for (int l = 0; l < NLAYER; ++l) {
    // params flattened per-layer in dict insertion order:
    // W, W_e, att_src, att_dst, att_edge, bias, ln_g, ln_b
    const int pb = 4 + l * 8;
    const float* W     = (const float*)d_in[pb + 0];
    const float* We    = (const float*)d_in[pb + 1];
    const float* att_s = (const float*)d_in[pb + 2];
    const float* att_d = (const float*)d_in[pb + 3];
    const float* att_e = (const float*)d_in[pb + 4];
    const float* bias  = (const float*)d_in[pb + 5];
    const float* ln_g  = (const float*)d_in[pb + 6];
    const float* ln_b  = (const float*)d_in[pb + 7];

    const float* Xin = (l == 0) ? x_in : xcur;
    const int K = (l == 0) ? EMB : HID;

    gemm_xs_kernel<<<NNODES / 16, 256, 0, stream>>>(Xin, W, xs, K);
    attn_node_kernel<<<(NNODES * HEADS + 255) / 256, 256, 0, stream>>>(xs, att_s, att_d, asrc, adst);
    ve_kernel<<<1, 128, 0, stream>>>(We, att_e, ve);

    fill_u32<<<(NNODES * HEADS + 255) / 256, 256, 0, stream>>>(amax, ENC_NEG_INF, NNODES * HEADS);
    fill_f32<<<(NNODES * HEADS + 255) / 256, 256, 0, stream>>>(den, 0.f, NNODES * HEADS);
    fill_f32<<<(NNODES * OUTC + 255) / 256, 256, 0, stream>>>(agg, 0.f, NNODES * OUTC);

    edge_pass1<<<(ETOT + 255) / 256, 256, 0, stream>>>(edge_attr, eamean, ve, ei, asrc, adst, sc, amax);
    edge_pass2<<<(ETOT * HEADS + 255) / 256, 256, 0, stream>>>(sc, ei, amax, den);
    scatter_kernel<<<((size_t)ETOT * 32 + 255) / 256, 256, 0, stream>>>(xs, sc, den, ei, agg);
    finalize_kernel<<<NNODES, 128, 0, stream>>>(agg, bias, ln_g, ln_b, xcur);
  }

  float* outp = (float*)d_out;
  fill_f32<<<(NGRAPH * HID + 255) / 256, 256, 0, stream>>>(outp, 0.f, NGRAPH * HID);
  fill_f32<<<(NGRAPH + 255) / 256, 256, 0, stream>>>(cnt, 0.f, NGRAPH);
  pool_accum<<<(NNODES * HID + 255) / 256, 256, 0, stream>>>(xcur, batch, outp, cnt);
  pool_div<<<(NGRAPH * HID + 255) / 256, 256, 0, stream>>>(outp, cnt);
}